// QNR_55559696941703
// MI455X (gfx1250) — compile-verified
//
#include <hip/hip_runtime.h>

typedef __attribute__((ext_vector_type(2))) float v2f;
typedef __attribute__((ext_vector_type(8))) float v8f;

#define REC 88   // per-block partial record: [0..63] gram, [64..71] sum_x, [72..79] sum_xp, [80] sum_p, [81] sum_pp

// ---------------------------------------------------------------------------
// Kernel 1: 4x4 mean-pool of pan (4,1,1024,1024) -> pan_l (4,1,256,256)
// ---------------------------------------------------------------------------
__global__ __launch_bounds__(256) void down4_kernel(const float* __restrict__ pan,
                                                    float* __restrict__ panl) {
  int o   = blockIdx.x * 256 + threadIdx.x;   // 262144 outputs
  int n   = o >> 16;
  int rem = o & 65535;
  int i = rem >> 8, j = rem & 255;
  const float* base = pan + (size_t)n * (1024u * 1024u) + (size_t)(i * 4) * 1024 + j * 4;
  float s = 0.f;
#pragma unroll
  for (int r = 0; r < 4; ++r) {
    float4 v = *(const float4*)(base + r * 1024);
    s += (v.x + v.y) + (v.z + v.w);
  }
  panl[o] = s * 0.0625f;
}

// ---------------------------------------------------------------------------
// Kernel 2/3: fused 8x8 channel Gram (via V_WMMA_F32_16X16X4_F32) + channel
// sums + pan cross-stats. x: (4,8,HW) fp32, p: (4,HW) fp32.
// grid = (blocksPerN, 4), block = 256 (8 waves).
// ---------------------------------------------------------------------------
__global__ __launch_bounds__(256) void gramstats_kernel(const float* __restrict__ x,
                                                        const float* __restrict__ p,
                                                        float* __restrict__ part,
                                                        int HW, int seg) {
  __shared__ float lds_g[8][32][8];
  __shared__ float lds_s[8][32][4];
  const int tid  = threadIdx.x;
  const int wave = tid >> 5, lane = tid & 31;
  const int n    = blockIdx.y;

  // WMMA A-matrix (16x4 f32) lane mapping: row m = lane%16, kk = 2*(lane/16)+r.
  // Rows 0-7  = channels 0-7 over K-chunk [base..base+3]
  // Rows 8-15 = channels 0-7 over K-chunk [base+4..base+7]
  const int c      = lane & 7;
  const int posOff = ((lane >> 3) & 1) * 4 + (lane >> 4) * 2;

  const float* xc = x + ((size_t)n * 8 + c) * (size_t)HW;
  const float* pn = p + (size_t)n * (size_t)HW;

  const int waveRun = seg >> 3;                       // positions per wave
  int pos = blockIdx.x * seg + wave * waveRun + posOff;
  const int iters = waveRun >> 3;                     // 8 positions per iter

  v8f acc = {};
  float sx = 0.f, sxp = 0.f, sp = 0.f, spp = 0.f;
  for (int it = 0; it < iters; ++it, pos += 8) {
    float2 xv = *(const float2*)(xc + pos);
    float2 pv = *(const float2*)(pn + pos);
    v2f a; a[0] = xv.x; a[1] = xv.y;
    // self-Gram: A == B; diagonal 8x8 blocks of D accumulate x[c]·x[d]
    acc = __builtin_amdgcn_wmma_f32_16x16x4_f32(false, a, false, a,
                                                (short)0, acc, false, false);
    sx  += xv.x + xv.y;
    sxp  = fmaf(xv.x, pv.x, fmaf(xv.y, pv.y, sxp));
    sp  += pv.x + pv.y;
    spp  = fmaf(pv.x, pv.x, fmaf(pv.y, pv.y, spp));
  }

#pragma unroll
  for (int v = 0; v < 8; ++v) lds_g[wave][lane][v] = acc[v];
  lds_s[wave][lane][0] = sx;
  lds_s[wave][lane][1] = sxp;
  lds_s[wave][lane][2] = sp;
  lds_s[wave][lane][3] = spp;
  __syncthreads();

  float* rec = part + (size_t)(blockIdx.y * gridDim.x + blockIdx.x) * REC;
  if (tid < 64) {
    // Gram[c][d] = D[c][d] + D[c+8][d+8]  (C/D layout: vgpr=m%8, lane=n+16*(m/8))
    int cc = tid >> 3, d = tid & 7;
    float g = 0.f;
    for (int w = 0; w < 8; ++w) g += lds_g[w][d][cc] + lds_g[w][24 + d][cc];
    rec[cc * 8 + d] = g;
  } else if (tid < 72) {
    int cc = tid - 64; float s = 0.f;      // lanes {c, c+8, c+16, c+24} cover all 8 positions once
    for (int w = 0; w < 8; ++w)
      for (int j = 0; j < 4; ++j) s += lds_s[w][cc + 8 * j][0];
    rec[64 + cc] = s;
  } else if (tid < 80) {
    int cc = tid - 72; float s = 0.f;
    for (int w = 0; w < 8; ++w)
      for (int j = 0; j < 4; ++j) s += lds_s[w][cc + 8 * j][1];
    rec[72 + cc] = s;
  } else if (tid == 80 || tid == 81) {
    int comp = tid - 78;                   // 2 -> sum_p, 3 -> sum_pp
    float s = 0.f;                         // every position counted 8x (once per channel group)
    for (int w = 0; w < 8; ++w)
      for (int l = 0; l < 32; ++l) s += lds_s[w][l][comp];
    rec[tid] = s * 0.125f;                 // exact /8
  }
}

// ---------------------------------------------------------------------------
// Final combine helpers (f64 scalar math, trivial cost)
// ---------------------------------------------------------------------------
__device__ inline void q_all_pairs_dev(const float acc[][82], double HW, double q[8][8]) {
  for (int c = 0; c < 8; ++c) for (int d = 0; d < 8; ++d) q[c][d] = 0.0;
  for (int n = 0; n < 4; ++n) {
    double E[8], var[8];
    for (int c = 0; c < 8; ++c) {
      E[c]   = acc[n][64 + c] / HW;
      var[c] = acc[n][c * 8 + c] / HW - E[c] * E[c];
    }
    for (int c = 0; c < 8; ++c)
      for (int d = 0; d < 8; ++d) {
        double Eab = acc[n][c * 8 + d] / HW;
        double cov = Eab - E[c] * E[d];
        q[c][d] += 4.0 * cov * E[c] * E[d] /
                   ((var[c] + var[d]) * (E[c] * E[c] + E[d] * E[d]) + 1e-8);
      }
  }
  for (int c = 0; c < 8; ++c) for (int d = 0; d < 8; ++d) q[c][d] *= 0.25;
}

__device__ inline void q_vs_pan_dev(const float acc[][82], double HW, double q[8]) {
  for (int c = 0; c < 8; ++c) q[c] = 0.0;
  for (int n = 0; n < 4; ++n) {
    double Ep = acc[n][80] / HW;
    double vp = acc[n][81] / HW - Ep * Ep;
    for (int c = 0; c < 8; ++c) {
      double E   = acc[n][64 + c] / HW;
      double var = acc[n][c * 8 + c] / HW - E * E;
      double Eab = acc[n][72 + c] / HW;
      double cov = Eab - E * Ep;
      q[c] += 4.0 * cov * E * Ep / ((var + vp) * (E * E + Ep * Ep) + 1e-8);
    }
  }
  for (int c = 0; c < 8; ++c) q[c] *= 0.25;
}

__global__ __launch_bounds__(128) void finalize_kernel(const float* __restrict__ oPart, int oBlocks,
                                                       const float* __restrict__ mPart, int mBlocks,
                                                       float* __restrict__ res) {
  __shared__ float accO[4][82];
  __shared__ float accM[4][82];
  int t = threadIdx.x;
  if (t < 82) {
    for (int n = 0; n < 4; ++n) {
      float a = 0.f;
      for (int b = 0; b < oBlocks; ++b) a += oPart[(size_t)(n * oBlocks + b) * REC + t];
      accO[n][t] = a;
      a = 0.f;
      for (int b = 0; b < mBlocks; ++b) a += mPart[(size_t)(n * mBlocks + b) * REC + t];
      accM[n][t] = a;
    }
  }
  __syncthreads();
  if (t == 0) {
    double qps[8][8], qms[8][8], qop[8], qmp[8];
    q_all_pairs_dev(accO, 1048576.0, qps);
    q_all_pairs_dev(accM, 65536.0, qms);
    q_vs_pan_dev(accO, 1048576.0, qop);
    q_vs_pan_dev(accM, 65536.0, qmp);
    double Dl = 0.0;
    for (int c = 0; c < 8; ++c)
      for (int d = 0; d < 8; ++d)
        if (c != d) Dl += fabs(qps[c][d] - qms[c][d]);
    Dl /= 56.0;
    double Ds = 0.0;
    for (int c = 0; c < 8; ++c) Ds += fabs(qop[c] - qmp[c]);
    Ds *= 0.125;
    res[0] = (float)((1.0 - Dl) * (1.0 - Ds));
    res[1] = (float)Dl;
    res[2] = (float)Ds;
  }
}

// ---------------------------------------------------------------------------
extern "C" void kernel_launch(void* const* d_in, const int* in_sizes, int n_in,
                              void* d_out, int out_size, void* d_ws, size_t ws_size,
                              hipStream_t stream) {
  const float* pan = (const float*)d_in[0];   // (4,1,1024,1024)
  const float* ms  = (const float*)d_in[1];   // (4,8,256,256)
  const float* out = (const float*)d_in[2];   // (4,8,1024,1024)

  float* wsf   = (float*)d_ws;
  float* panl  = wsf;                          // 4*256*256 floats
  float* oPart = wsf + 4 * 256 * 256;          // 4*128 records
  float* mPart = oPart + 4 * 128 * REC;        // 4*16 records

  down4_kernel<<<1024, 256, 0, stream>>>(pan, panl);
  gramstats_kernel<<<dim3(128, 4), 256, 0, stream>>>(out, pan, oPart, 1024 * 1024, 8192);
  gramstats_kernel<<<dim3(16, 4), 256, 0, stream>>>(ms, panl, mPart, 256 * 256, 4096);
  finalize_kernel<<<1, 128, 0, stream>>>(oPart, 128, mPart, 16, (float*)d_out);
}